// GPSLayer_74010876444908
// MI455X (gfx1250) — compile-verified
//
#include <hip/hip_runtime.h>
#include <hip/hip_bf16.h>
#include <math.h>

// ---------------- problem constants (match reference) ----------------
#define BB   16
#define NN   512
#define DD   256
#define HH   8
#define DH   32
#define NT   (BB * NN)        // 8192 total nodes
#define BN_EPS 1e-5f

typedef float v2f __attribute__((ext_vector_type(2)));
typedef float v8f __attribute__((ext_vector_type(8)));

// f32 WMMA: D(16x16) = A(16x4) * B(4x16) + C   -- v_wmma_f32_16x16x4_f32
static __device__ __forceinline__ v8f wmma4(v2f a, v2f b, v8f c) {
    return __builtin_amdgcn_wmma_f32_16x16x4_f32(
        /*neg_a=*/false, a, /*neg_b=*/false, b,
        /*c_mod=*/(short)0, c, /*reuse_a=*/false, /*reuse_b=*/false);
}

static __device__ __forceinline__ float gelu_exact(float v) {
    return 0.5f * v * (1.0f + erff(v * 0.70710678118654752f));
}

// ====================================================================
// Register-blocked GEMM: C[M,N] = A[M,K] @ W[N,K]^T (+bias)(+GELU)
// One wave per 32x64 C block (2 M-tiles x 4 N-tiles).
// Per K-step of 4: 6 x b64 loads feed 8 WMMAs (A reused 4x, B reused 2x).
// ====================================================================
__global__ __launch_bounds__(256)
void gemm_wmma_f32(const float* __restrict__ A, const float* __restrict__ W,
                   const float* __restrict__ bias, float* __restrict__ C,
                   int M, int N, int K, int fuse_gelu)
{
    const int wave = blockIdx.x * (blockDim.x >> 5) + (threadIdx.x >> 5);
    const int ntn  = N >> 6;                 // 64-wide N blocks
    const int tm   = wave / ntn;
    const int tn   = wave - tm * ntn;
    if (tm >= (M >> 5)) return;

    const int lane = threadIdx.x & 31;
    const int half = lane >> 4;              // K-pair selector within step of 4
    const int lm   = lane & 15;              // M index (A) / N index (B)

    const int m0 = tm * 32;
    const int n0 = tn * 64;

    const float* Arow0 = A + (size_t)(m0 + lm) * K;
    const float* Arow1 = A + (size_t)(m0 + 16 + lm) * K;
    const float* Wr0   = W + (size_t)(n0 + 0 * 16 + lm) * K;
    const float* Wr1   = W + (size_t)(n0 + 1 * 16 + lm) * K;
    const float* Wr2   = W + (size_t)(n0 + 2 * 16 + lm) * K;
    const float* Wr3   = W + (size_t)(n0 + 3 * 16 + lm) * K;

    v8f acc[2][4];
#pragma unroll
    for (int i = 0; i < 2; i++)
#pragma unroll
        for (int j = 0; j < 4; j++) acc[i][j] = (v8f){};

#pragma unroll 2
    for (int k = 0; k < K; k += 4) {
        const int ko = k + 2 * half;
        v2f a0 = *(const v2f*)(Arow0 + ko);
        v2f a1 = *(const v2f*)(Arow1 + ko);
        v2f b0 = *(const v2f*)(Wr0 + ko);
        v2f b1 = *(const v2f*)(Wr1 + ko);
        v2f b2 = *(const v2f*)(Wr2 + ko);
        v2f b3 = *(const v2f*)(Wr3 + ko);
        acc[0][0] = wmma4(a0, b0, acc[0][0]);
        acc[0][1] = wmma4(a0, b1, acc[0][1]);
        acc[0][2] = wmma4(a0, b2, acc[0][2]);
        acc[0][3] = wmma4(a0, b3, acc[0][3]);
        acc[1][0] = wmma4(a1, b0, acc[1][0]);
        acc[1][1] = wmma4(a1, b1, acc[1][1]);
        acc[1][2] = wmma4(a1, b2, acc[1][2]);
        acc[1][3] = wmma4(a1, b3, acc[1][3]);
    }

    float bv[4];
#pragma unroll
    for (int j = 0; j < 4; j++)
        bv[j] = bias ? bias[n0 + j * 16 + lm] : 0.0f;

#pragma unroll
    for (int i = 0; i < 2; i++) {
#pragma unroll
        for (int r = 0; r < 8; r++) {
            const size_t row = (size_t)(m0 + i * 16 + r + 8 * half) * N;
#pragma unroll
            for (int j = 0; j < 4; j++) {
                float v = acc[i][j][r] + bv[j];
                if (fuse_gelu) v = gelu_exact(v);
                C[row + n0 + j * 16 + lm] = v;
            }
        }
    }
}

// ====================================================================
// Sparse scatter: agg[row] += val * hl[col]  (32 lanes/edge, 8 ch each)
// ====================================================================
__global__ __launch_bounds__(256)
void scatter_edges(const int* __restrict__ rows, const int* __restrict__ cols,
                   const float* __restrict__ vals, const float* __restrict__ hl,
                   float* __restrict__ agg, int E)
{
    int t = blockIdx.x * blockDim.x + threadIdx.x;
    int e = t >> 5;
    if (e >= E) return;
    int c0 = (t & 31) * 8;
    int r = rows[e], c = cols[e];
    float v = vals[e];
    const float4* src = (const float4*)(hl + (size_t)c * DD + c0);
    float* dst = agg + (size_t)r * DD + c0;
    float4 h0 = src[0], h1 = src[1];
    atomicAdd(dst + 0, v * h0.x); atomicAdd(dst + 1, v * h0.y);
    atomicAdd(dst + 2, v * h0.z); atomicAdd(dst + 3, v * h0.w);
    atomicAdd(dst + 4, v * h1.x); atomicAdd(dst + 5, v * h1.y);
    atomicAdd(dst + 6, v * h1.z); atomicAdd(dst + 7, v * h1.w);
}

__global__ __launch_bounds__(256)
void zero_f32(float* __restrict__ p, int n)
{
    int i = blockIdx.x * blockDim.x + threadIdx.x;
    if (i < n) p[i] = 0.0f;
}

// ====================================================================
// Elementwise epilogues: residual + (optional GELU on addend) + BN(eval)
// ====================================================================
__global__ __launch_bounds__(256)
void add_gelu_bn(const float* __restrict__ x, const float* __restrict__ agg,
                 const float* __restrict__ g, const float* __restrict__ b,
                 const float* __restrict__ m, const float* __restrict__ v,
                 float* __restrict__ out, int total)
{
    int i = blockIdx.x * blockDim.x + threadIdx.x;
    if (i >= total) return;
    int c = i & (DD - 1);
    float val = x[i] + gelu_exact(agg[i]);
    out[i] = (val - m[c]) * (g[c] * rsqrtf(v[c] + BN_EPS)) + b[c];
}

__global__ __launch_bounds__(256)
void add_bn(const float* __restrict__ a, const float* __restrict__ bsrc,
            const float* __restrict__ g, const float* __restrict__ b,
            const float* __restrict__ m, const float* __restrict__ v,
            float* __restrict__ out, int total)
{
    int i = blockIdx.x * blockDim.x + threadIdx.x;
    if (i >= total) return;
    int c = i & (DD - 1);
    float val = a[i] + bsrc[i];
    out[i] = (val - m[c]) * (g[c] * rsqrtf(v[c] + BN_EPS)) + b[c];
}

// ====================================================================
// Flash-style attention, one wave per (b, h, 16-query tile).
// qkv layout: [NT, 768] with q at col 0, k at col 256, v at col 512;
// head h occupies cols h*32 .. h*32+31 within each third.
// ctx out: [NT, 256] with head h at cols h*32..h*32+31.
// ====================================================================
__global__ __launch_bounds__(256)
void attn_wmma(const float* __restrict__ qkv, float* __restrict__ ctx)
{
    __shared__ float pbuf[8][16 * 16];
    const int warp = threadIdx.x >> 5;
    const int gid  = blockIdx.x * 8 + warp;       // over B*H*(N/16) = 4096
    const int qt   = gid & 31;
    const int h    = (gid >> 5) & 7;
    const int b    = gid >> 8;
    const int lane = threadIdx.x & 31;
    const int half = lane >> 4;
    const int lm   = lane & 15;
    const int LDQ  = 3 * DD;                      // 768

    const float* gbase = qkv + (size_t)(b * NN) * LDQ;

    // Preload Q A-fragments (16x32 tile -> 8 K-steps of 4)
    v2f qa[8];
    {
        const float* qrow = gbase + (size_t)(qt * 16 + lm) * LDQ + h * DH;
#pragma unroll
        for (int j = 0; j < 8; j++) qa[j] = *(const v2f*)(qrow + 4 * j + 2 * half);
    }

    float rowmax[8], rowsum[8];
    v8f o0 = {}, o1 = {};
#pragma unroll
    for (int r = 0; r < 8; r++) { rowmax[r] = -1e30f; rowsum[r] = 0.0f; }
    const float scale = 0.17677669529663687f;     // 1/sqrt(32)
    float* pb = pbuf[warp];

    for (int kt = 0; kt < NN / 16; kt++) {
        // ---- S = Q @ Ktile^T  (16x16) ----
        v8f s = {};
        const float* krow = gbase + (size_t)(kt * 16 + lm) * LDQ + DD + h * DH;
#pragma unroll
        for (int j = 0; j < 8; j++) {
            v2f kb = *(const v2f*)(krow + 4 * j + 2 * half);
            s = wmma4(qa[j], kb, s);
        }

        // ---- online softmax (row-wise over 16-lane halves) ----
        float alpha[8];
#pragma unroll
        for (int r = 0; r < 8; r++) {
            float val = s[r] * scale;
            float mx = val;
            mx = fmaxf(mx, __shfl_xor(mx, 1));
            mx = fmaxf(mx, __shfl_xor(mx, 2));
            mx = fmaxf(mx, __shfl_xor(mx, 4));
            mx = fmaxf(mx, __shfl_xor(mx, 8));
            float nm = fmaxf(rowmax[r], mx);
            float p  = __expf(val - nm);
            float sm = p;
            sm += __shfl_xor(sm, 1);
            sm += __shfl_xor(sm, 2);
            sm += __shfl_xor(sm, 4);
            sm += __shfl_xor(sm, 8);
            float al = __expf(rowmax[r] - nm);
            rowmax[r] = nm;
            rowsum[r] = rowsum[r] * al + sm;
            alpha[r]  = al;
            s[r] = p;
        }
#pragma unroll
        for (int r = 0; r < 8; r++) { o0[r] *= alpha[r]; o1[r] *= alpha[r]; }

        // ---- stage P (C layout) into LDS, re-read as A fragments ----
#pragma unroll
        for (int r = 0; r < 8; r++) pb[(r + 8 * half) * 16 + lm] = s[r];
        asm volatile("s_wait_dscnt 0" ::: "memory");   // wave-local RAW fence

        // ---- O += P @ Vtile  (K=16 -> 4 steps; dh=32 -> 2 N-halves) ----
#pragma unroll
        for (int j = 0; j < 4; j++) {
            v2f pa = *(const v2f*)(pb + lm * 16 + 4 * j + 2 * half);
            const float* vp = gbase + (size_t)(kt * 16 + 4 * j + 2 * half) * LDQ
                              + 2 * DD + h * DH;
            v2f vb0, vb1;
            vb0.x = vp[lm];            vb0.y = vp[LDQ + lm];
            vb1.x = vp[16 + lm];       vb1.y = vp[LDQ + 16 + lm];
            o0 = wmma4(pa, vb0, o0);
            o1 = wmma4(pa, vb1, o1);
        }
        asm volatile("s_wait_dscnt 0" ::: "memory");   // loads done before reuse
    }

    // ---- normalize and store ctx ----
#pragma unroll
    for (int r = 0; r < 8; r++) {
        float inv = 1.0f / rowsum[r];
        int row = b * NN + qt * 16 + r + 8 * half;
        ctx[(size_t)row * DD + h * DH + lm]      = o0[r] * inv;
        ctx[(size_t)row * DD + h * DH + 16 + lm] = o1[r] * inv;
    }
}

// ====================================================================
// Host-side orchestration
// ====================================================================
static inline int gemm_blocks(int M, int N) { return ((M >> 5) * (N >> 6)) / 8; }

extern "C" void kernel_launch(void* const* d_in, const int* in_sizes, int n_in,
                              void* d_out, int out_size, void* d_ws, size_t ws_size,
                              hipStream_t stream)
{
    (void)n_in; (void)out_size; (void)ws_size;
    const float* x          = (const float*)d_in[0];
    const int*   edge_rows  = (const int*)  d_in[1];
    const int*   edge_cols  = (const int*)  d_in[2];
    const float* edge_vals  = (const float*)d_in[3];
    const float* w_gcn      = (const float*)d_in[4];
    const float* in_proj_w  = (const float*)d_in[5];
    const float* in_proj_b  = (const float*)d_in[6];
    const float* out_proj_w = (const float*)d_in[7];
    const float* out_proj_b = (const float*)d_in[8];
    const float* w1         = (const float*)d_in[9];
    const float* b1         = (const float*)d_in[10];
    const float* w2         = (const float*)d_in[11];
    const float* b2         = (const float*)d_in[12];
    const float* bn1_g = (const float*)d_in[15], *bn1_b = (const float*)d_in[16];
    const float* bn1_m = (const float*)d_in[17], *bn1_v = (const float*)d_in[18];
    const float* bn2_g = (const float*)d_in[19], *bn2_b = (const float*)d_in[20];
    const float* bn2_m = (const float*)d_in[21], *bn2_v = (const float*)d_in[22];
    const float* bn3_g = (const float*)d_in[23], *bn3_b = (const float*)d_in[24];
    const float* bn3_m = (const float*)d_in[25], *bn3_v = (const float*)d_in[26];

    const int E = in_sizes[1];          // number of edges

    float* ws = (float*)d_ws;
    float* wsA = ws;                                  // 8192 x 1024 : qkv / h1
    float* wsB = wsA + (size_t)NT * 1024;             // 8192 x 256  : hl  / ctx
    float* wsC = wsB + (size_t)NT * DD;               // 8192 x 256  : agg / attnout / ffn2
    float* wsD = wsC + (size_t)NT * DD;               // 8192 x 256  : x1
    float* wsE = wsD + (size_t)NT * DD;               // 8192 x 256  : x2
    float* outp = (float*)d_out;

    const int EW = NT * DD;                           // 2,097,152 elements
    const dim3 blk(256);

    // 1) hl = x @ w_gcn^T
    gemm_wmma_f32<<<gemm_blocks(NT, DD), blk, 0, stream>>>(x, w_gcn, nullptr, wsB,
                                                           NT, DD, DD, 0);
    // 2) agg = 0 ; agg[row] += val * hl[col]
    zero_f32<<<(EW + 255) / 256, blk, 0, stream>>>(wsC, EW);
    scatter_edges<<<(E * 32 + 255) / 256, blk, 0, stream>>>(edge_rows, edge_cols,
                                                            edge_vals, wsB, wsC, E);
    // 3) x1 = bn1(x + gelu(agg))
    add_gelu_bn<<<(EW + 255) / 256, blk, 0, stream>>>(x, wsC, bn1_g, bn1_b, bn1_m,
                                                      bn1_v, wsD, EW);
    // 4) qkv = x1 @ in_proj_w^T + in_proj_b
    gemm_wmma_f32<<<gemm_blocks(NT, 3 * DD), blk, 0, stream>>>(wsD, in_proj_w,
                                                               in_proj_b, wsA,
                                                               NT, 3 * DD, DD, 0);
    // 5) ctx = multihead attention(qkv)
    attn_wmma<<<(BB * HH * (NN / 16)) / 8, blk, 0, stream>>>(wsA, wsB);
    // 6) attnout = ctx @ out_proj_w^T + out_proj_b
    gemm_wmma_f32<<<gemm_blocks(NT, DD), blk, 0, stream>>>(wsB, out_proj_w,
                                                           out_proj_b, wsC,
                                                           NT, DD, DD, 0);
    // 7) x2 = bn2(x1 + attnout)
    add_bn<<<(EW + 255) / 256, blk, 0, stream>>>(wsD, wsC, bn2_g, bn2_b, bn2_m,
                                                 bn2_v, wsE, EW);
    // 8) h1 = gelu(x2 @ w1^T + b1)   (GELU fused into GEMM epilogue)
    gemm_wmma_f32<<<gemm_blocks(NT, 4 * DD), blk, 0, stream>>>(wsE, w1, b1, wsA,
                                                               NT, 4 * DD, DD, 1);
    // 9) ffn2 = h1 @ w2^T + b2
    gemm_wmma_f32<<<gemm_blocks(NT, DD), blk, 0, stream>>>(wsA, w2, b2, wsC,
                                                           NT, DD, 4 * DD, 0);
    // 10) out = bn3(x2 + ffn2)
    add_bn<<<(EW + 255) / 256, blk, 0, stream>>>(wsE, wsC, bn3_g, bn3_b, bn3_m,
                                                 bn3_v, outp, EW);
}